// GCNGraphEmbedder_65687229825055
// MI455X (gfx1250) — compile-verified
//
#include <hip/hip_runtime.h>
#include <math.h>

// GCN 2-layer forward for MI455X (gfx1250).
// Dense transforms use V_WMMA_F32_16X16X4_F32 (exact fp32 matrix pipe).
// Message passing = coalesced gather + global fp32 atomics (L2-resident buffers).

typedef __attribute__((ext_vector_type(2))) float v2f;
typedef __attribute__((ext_vector_type(8))) float v8f;

#define CH 64  // both hidden and output channel width

// ---------------- elementwise / fill ----------------
__global__ void k_fill(float* __restrict__ p, float v, int n) {
  int i = blockIdx.x * blockDim.x + threadIdx.x;
  if (i < n) p[i] = v;
}

// deg[dst] += 1 per edge (deg buffer pre-filled with 1.0 = self loop)
__global__ void k_edge_deg(const long long* __restrict__ dst, float* __restrict__ deg, int E) {
  int e = blockIdx.x * blockDim.x + threadIdx.x;
  if (e < E) atomicAdd(&deg[(int)dst[e]], 1.0f);
}

// in-place deg -> deg^{-1/2}  (deg >= 1 always, self loops)
__global__ void k_dinv(float* __restrict__ d, int n) {
  int i = blockIdx.x * blockDim.x + threadIdx.x;
  if (i < n) d[i] = 1.0f / sqrtf(d[i]);
}

// ---------------- fp32 WMMA GEMM:  Y[M,64] = A[M,K] @ W[K,64] ----------------
// grid.x = M/16 (M multiple of 16), block = 128 threads = 4 waves,
// wave w computes the 16x16 tile at (blockIdx.x*16, w*16).
__global__ void k_gemm_wmma64(const float* __restrict__ A, const float* __restrict__ W,
                              float* __restrict__ Y, int K) {
  const int lane = threadIdx.x & 31;
  const int wave = threadIdx.x >> 5;
  const int row0 = blockIdx.x * 16;
  const int col0 = wave * 16;
  const int mn   = lane & 15;          // row (A) / col (B,D) within tile
  const int kh   = (lane >> 4) << 1;   // K sub-offset per lane half: 0 or 2

  v8f acc = {};
  const float* arow = A + (size_t)(row0 + mn) * K + kh;        // A[row][kh + k + {0,1}]
  const float* bcol = W + (size_t)kh * CH + col0 + mn;         // W[kh + k + {0,1}][col]

  for (int k = 0; k < K; k += 4) {
    v2f a, b;
    a.x = arow[k];
    a.y = arow[k + 1];
    b.x = bcol[(size_t)k * CH];
    b.y = bcol[(size_t)(k + 1) * CH];
    // D = A(16x4) * B(4x16) + C   — v_wmma_f32_16x16x4_f32
    acc = __builtin_amdgcn_wmma_f32_16x16x4_f32(
        /*neg_a=*/false, a, /*neg_b=*/false, b,
        /*c_mod=*/(short)0, acc, /*reuse_a=*/false, /*reuse_b=*/false);
  }

  // D layout: VGPR r -> row r + (lane>=16 ? 8 : 0), col = lane&15
  float* yrow = Y + (size_t)(row0 + ((lane >> 4) << 3)) * CH + col0 + mn;
#pragma unroll
  for (int r = 0; r < 8; ++r) yrow[(size_t)r * CH] = acc[r];
}

// ---------------- message passing ----------------
// self-loop term initializes accumulator: Acc[n][c] = X[n][c] * dinv[n]^2
__global__ void k_selfloop(const float* __restrict__ X, const float* __restrict__ dinv,
                           float* __restrict__ Acc, int total) {
  int i = blockIdx.x * blockDim.x + threadIdx.x;
  if (i >= total) return;
  float di = dinv[i >> 6];
  Acc[i] = X[i] * di * di;
}

// Acc[dst][c] += X[src][c] * dinv[src]*dinv[dst]   (64 lanes per edge, coalesced)
__global__ void k_edge_scatter(const long long* __restrict__ src, const long long* __restrict__ dst,
                               const float* __restrict__ dinv, const float* __restrict__ X,
                               float* __restrict__ Acc, int E) {
  long long t = (long long)blockIdx.x * blockDim.x + threadIdx.x;
  if (t >= (long long)E * CH) return;
  int e = (int)(t >> 6);
  int c = (int)(t & 63);
  int s = (int)src[e];
  int d = (int)dst[e];
  float w = dinv[s] * dinv[d];
  atomicAdd(&Acc[(size_t)d * CH + c], X[(size_t)s * CH + c] * w);
}

// H[i] = relu(Acc[i] + b[c])
__global__ void k_bias_relu(const float* __restrict__ Acc, const float* __restrict__ b,
                            float* __restrict__ H, int total) {
  int i = blockIdx.x * blockDim.x + threadIdx.x;
  if (i >= total) return;
  H[i] = fmaxf(Acc[i] + b[i & 63], 0.0f);
}

// out[batch[n]][c] += Acc[n][c] + b2[c];  cnt[batch[n]] += 1 (from c==0 lanes)
__global__ void k_pool(const float* __restrict__ Acc, const float* __restrict__ b2,
                       const long long* __restrict__ batch,
                       float* __restrict__ out, float* __restrict__ cnt, int total) {
  int i = blockIdx.x * blockDim.x + threadIdx.x;
  if (i >= total) return;
  int n = i >> 6;
  int c = i & 63;
  int g = (int)batch[n];
  atomicAdd(&out[(size_t)g * CH + c], Acc[i] + b2[c]);
  if (c == 0) atomicAdd(&cnt[g], 1.0f);
}

__global__ void k_pool_div(float* __restrict__ out, const float* __restrict__ cnt, int total) {
  int i = blockIdx.x * blockDim.x + threadIdx.x;
  if (i >= total) return;
  out[i] /= fmaxf(cnt[i >> 6], 1.0f);
}

// ---------------- launch ----------------
extern "C" void kernel_launch(void* const* d_in, const int* in_sizes, int n_in,
                              void* d_out, int out_size, void* d_ws, size_t ws_size,
                              hipStream_t stream) {
  // setup_inputs() order: x, W1, b1, W2, b2, edge_index, batch, num_graphs
  const float*     x     = (const float*)d_in[0];
  const float*     W1    = (const float*)d_in[1];
  const float*     b1    = (const float*)d_in[2];
  const float*     W2    = (const float*)d_in[3];
  const float*     b2    = (const float*)d_in[4];
  const long long* ei    = (const long long*)d_in[5];  // int64 [2, E]
  const long long* batch = (const long long*)d_in[6];  // int64 [N]

  const int INCH = 128;
  const int N = in_sizes[0] / INCH;   // 50000 (multiple of 16)
  const int E = in_sizes[5] / 2;      // 800000
  const int G = out_size / CH;        // 512

  const long long* src = ei;
  const long long* dst = ei + E;
  float* out = (float*)d_out;

  // workspace: dinv | bufA | bufB | bufC | counts   (~39 MB)
  float* ws   = (float*)d_ws;
  size_t doff = ((size_t)N + 255) & ~(size_t)255;
  size_t fsz  = (size_t)N * CH;       // 3.2M floats per feature buffer
  float* dinv = ws;
  float* bufA = ws + doff;
  float* bufB = bufA + fsz;
  float* bufC = bufB + fsz;
  float* cnts = bufC + fsz;

  const int T = 256;
  auto cdiv = [](long long a, long long b) { return (int)((a + b - 1) / b); };

  // degrees (with self loop) -> dinv
  k_fill<<<cdiv(N, T), T, 0, stream>>>(dinv, 1.0f, N);
  k_edge_deg<<<cdiv(E, T), T, 0, stream>>>(dst, dinv, E);
  k_dinv<<<cdiv(N, T), T, 0, stream>>>(dinv, N);

  // layer 1: xl1 = x @ W1 (WMMA f32), then normalized aggregation, bias+relu
  k_gemm_wmma64<<<N / 16, 128, 0, stream>>>(x, W1, bufA, INCH);
  k_selfloop<<<cdiv((long long)fsz, T), T, 0, stream>>>(bufA, dinv, bufB, (int)fsz);
  k_edge_scatter<<<cdiv((long long)E * CH, T), T, 0, stream>>>(src, dst, dinv, bufA, bufB, E);
  k_bias_relu<<<cdiv((long long)fsz, T), T, 0, stream>>>(bufB, b1, bufA, (int)fsz);

  // layer 2: xl2 = h @ W2 (WMMA f32), then normalized aggregation
  k_gemm_wmma64<<<N / 16, 128, 0, stream>>>(bufA, W2, bufC, CH);
  k_selfloop<<<cdiv((long long)fsz, T), T, 0, stream>>>(bufC, dinv, bufB, (int)fsz);
  k_edge_scatter<<<cdiv((long long)E * CH, T), T, 0, stream>>>(src, dst, dinv, bufC, bufB, E);

  // global mean pool (bias b2 folded in)
  k_fill<<<cdiv(G * CH, T), T, 0, stream>>>(out, 0.0f, G * CH);
  k_fill<<<cdiv(G, T), T, 0, stream>>>(cnts, 0.0f, G);
  k_pool<<<cdiv((long long)fsz, T), T, 0, stream>>>(bufB, b2, batch, out, cnts, (int)fsz);
  k_pool_div<<<cdiv(G * CH, T), T, 0, stream>>>(out, cnts, G * CH);
}